// ContrastiveLoss_CosineSimilarity_83863531421852
// MI455X (gfx1250) — compile-verified
//
#include <hip/hip_runtime.h>
#include <math.h>

// Problem constants (match the reference).
#define NROWS 8192
#define DDIM  1024
#define CH    4              // K-chunks per row (split-K for wave parallelism)
#define KC    (DDIM / CH)    // 256 floats per chunk

typedef __attribute__((ext_vector_type(2))) float v2f;
typedef __attribute__((ext_vector_type(8))) float v8f;

// Full-precision fp32 WMMA: D(16x16) = A(16x4) * B(4x16) + C
__device__ __forceinline__ v8f wmma4(v2f a, v2f b, v8f c) {
    // 8-arg pattern: (neg_a, A, neg_b, B, c_mod, C, reuse_a, reuse_b)
    return __builtin_amdgcn_wmma_f32_16x16x4_f32(
        false, a, false, b, (short)0, c, false, false);
}

// ---------------------------------------------------------------------------
// Phase 1: one wave per (16-row tile, K-chunk).
// Accumulates dots = diag(A*B^T), na = diag(A*A^T), nb = diag(B*B^T)
// for its tile/chunk via v_wmma_f32_16x16x4_f32, writes per-row partials.
// Grid: 256 blocks x 256 threads = 2048 waves = 512 tiles * 4 chunks.
// ---------------------------------------------------------------------------
__global__ void cosine_phase1(const float* __restrict__ A,
                              const float* __restrict__ B,
                              float* __restrict__ ws) {
    const int lane  = threadIdx.x & 31;
    const int wave  = (blockIdx.x * blockDim.x + threadIdx.x) >> 5; // 0..2047
    const int tile  = wave / CH;        // 0..511
    const int chunk = wave % CH;        // 0..3
    const int row0  = tile * 16;
    const int k0    = chunk * KC;

    // f32 16x4 A-operand layout: lanes 0-15 hold (M=lane, K=0..1) in v0/v1,
    // lanes 16-31 hold (M=lane-16, K=2..3)  -> contiguous b64 per lane.
    const int r    = lane & 15;
    const int koff = (lane < 16) ? 0 : 2;
    const float* pa = A + (size_t)(row0 + r) * DDIM + k0 + koff;
    const float* pb = B + (size_t)(row0 + r) * DDIM + k0 + koff;

    v8f dots = {}, na = {}, nb = {};
#pragma unroll 4
    for (int k = 0; k < KC; k += 4) {
        v2f a = *(const v2f*)(pa + k);
        v2f b = *(const v2f*)(pb + k);
        dots = wmma4(a, b, dots);   // diag -> row dot products
        na   = wmma4(a, a, na);     // diag -> ||a||^2
        nb   = wmma4(b, b, nb);     // diag -> ||b||^2
    }

    // Diagonal of 16x16 f32 D: VGPR m (m<8) at lane m; VGPR m-8 (m>=8) at
    // lane m+16 (i.e. lanes 24..31 hold rows 8..15 in acc[lane-24]).
    const bool have = (lane < 8) || (lane >= 24);
    const int  idx  = (lane < 8) ? lane : (lane - 24);
    const int  row  = (lane < 8) ? lane : (lane - 16);

    float d = 0.f, a2 = 0.f, b2 = 0.f;
#pragma unroll
    for (int i = 0; i < 8; ++i)
        if (idx == i) { d = dots[i]; a2 = na[i]; b2 = nb[i]; }

    if (have) {
        const int gr = row0 + row;
        ws[0 * NROWS * CH + gr * CH + chunk] = d;
        ws[1 * NROWS * CH + gr * CH + chunk] = a2;
        ws[2 * NROWS * CH + gr * CH + chunk] = b2;
    }
}

// ---------------------------------------------------------------------------
// Phase 2: per-row combine of K-chunk partials -> cosine score -> squared
// error; deterministic block reduction. Grid: 32 blocks x 256 threads.
// ---------------------------------------------------------------------------
__global__ void cosine_phase2(const float* __restrict__ ws,
                              const float* __restrict__ labels,
                              float* __restrict__ red) {
    const int gr = blockIdx.x * blockDim.x + threadIdx.x;   // 0..8191
    const float* wd = ws;
    const float* wa = ws + 1 * NROWS * CH;
    const float* wb = ws + 2 * NROWS * CH;

    float dot = 0.f, a2 = 0.f, b2 = 0.f;
#pragma unroll
    for (int c = 0; c < CH; ++c) {
        dot += wd[gr * CH + c];
        a2  += wa[gr * CH + c];
        b2  += wb[gr * CH + c];
    }
    const float score = dot / (sqrtf(a2) * sqrtf(b2));
    const float diff  = score - labels[gr];
    float sq = diff * diff;

    // wave32 shfl-xor tree, then cross-wave via LDS
    __shared__ float sdata[8];
#pragma unroll
    for (int off = 16; off > 0; off >>= 1)
        sq += __shfl_xor(sq, off, 32);
    const int lane = threadIdx.x & 31;
    const int wid  = threadIdx.x >> 5;
    if (lane == 0) sdata[wid] = sq;
    __syncthreads();
    if (wid == 0) {
        float v = (lane < 8) ? sdata[lane] : 0.f;
#pragma unroll
        for (int off = 4; off > 0; off >>= 1)
            v += __shfl_xor(v, off, 32);
        if (lane == 0) red[blockIdx.x] = v;
    }
}

// ---------------------------------------------------------------------------
// Phase 3: final 32-way reduce, mean, write scalar.
// ---------------------------------------------------------------------------
__global__ void cosine_phase3(const float* __restrict__ red,
                              float* __restrict__ out) {
    const int lane = threadIdx.x;           // 32 threads, 1 block
    float v = red[lane];
#pragma unroll
    for (int off = 16; off > 0; off >>= 1)
        v += __shfl_xor(v, off, 32);
    if (lane == 0) out[0] = v * (1.0f / (float)NROWS);
}

extern "C" void kernel_launch(void* const* d_in, const int* in_sizes, int n_in,
                              void* d_out, int out_size, void* d_ws, size_t ws_size,
                              hipStream_t stream) {
    (void)in_sizes; (void)n_in; (void)out_size; (void)ws_size;
    const float* A      = (const float*)d_in[0];   // issues_1_geb [N,D] f32
    const float* B      = (const float*)d_in[1];   // issues_2_geb [N,D] f32
    const float* labels = (const float*)d_in[2];   // labels [N] f32
    float*       out    = (float*)d_out;

    float* ws  = (float*)d_ws;                     // 3*N*CH partials
    float* red = ws + 3 * NROWS * CH;              // 32 block partials

    // 2048 waves: 512 16-row tiles x 4 K-chunks; each wave streams 32 KB.
    cosine_phase1<<<(NROWS / 16) * CH * 32 / 256, 256, 0, stream>>>(A, B, ws);
    cosine_phase2<<<NROWS / 256, 256, 0, stream>>>(ws, labels, red);
    cosine_phase3<<<1, 32, 0, stream>>>(red, out);
}